// GNN_62423054680439
// MI455X (gfx1250) — compile-verified
//
#include <hip/hip_runtime.h>
#include <hip/hip_bf16.h>
#include <math.h>

#define N_NODES   100000
#define N_EDGES   1600000
#define E_TOT     (N_EDGES + N_NODES)   // reference appends self-loops
#define N_GRAPHS  256
#define NEG_SLOPE 0.2f
#define EPS_SM    1e-16f
#define NEG_INF_F __int_as_float(0xFF800000)

typedef __attribute__((ext_vector_type(2))) float v2f;
typedef __attribute__((ext_vector_type(8))) float v8f;

// ---------- ordered-int float atomic max (exact; amax initialized to -inf) ----------
__device__ __forceinline__ void atomicMaxFloat(float* addr, float val) {
  if (val >= 0.0f) {
    atomicMax((int*)addr, __float_as_int(val));
  } else {
    atomicMin((unsigned int*)addr, __float_as_uint(val));
  }
}

__device__ __forceinline__ void edge_endpoints(const int* __restrict__ ei, int e, int& s, int& d) {
  if (e < N_EDGES) { s = ei[e]; d = ei[N_EDGES + e]; }
  else             { s = e - N_EDGES; d = s; }          // self-loop
}

// ---------- init all scratch (harness poisons ws; we must own every byte we read) ----------
__global__ __launch_bounds__(256) void gat_init(float* amax1, float* denom1, float* numer1,
                                                float* amax2, float* denom2, float* numer2,
                                                float* pooled, float* cnt) {
  int i = blockIdx.x * blockDim.x + threadIdx.x;        // covers N_NODES*64
  if (i < N_NODES) {
    amax1[i] = NEG_INF_F; denom1[i] = 0.f; numer1[i] = 0.f;
    amax2[i] = NEG_INF_F; denom2[i] = 0.f;
  }
  if (i < N_NODES * 64) numer2[i] = 0.f;
  if (i < N_GRAPHS * 64) pooled[i] = 0.f;
  if (i < N_GRAPHS) cnt[i] = 0.f;
}

// ---------- layer-1 constants: cs = W1·a1s, cd = W1·a1d (Fin==1 collapses attention) ----------
__global__ void gat_prep(const float* __restrict__ W1, const float* __restrict__ a1s,
                         const float* __restrict__ a1d, float* consts) {
  if (threadIdx.x == 0 && blockIdx.x == 0) {
    float cs = 0.f, cd = 0.f;
    for (int f = 0; f < 32; ++f) { cs += W1[f] * a1s[f]; cd += W1[f] * a1d[f]; }
    consts[0] = cs; consts[1] = cd;
  }
}

// ---------- layer 1, pass A: alpha = lrelu(cs*x[src] + cd*x[dst]); segment max ----------
__global__ __launch_bounds__(256) void l1_edgeA(const float* __restrict__ x, const int* __restrict__ ei,
                                                const float* __restrict__ consts,
                                                float* __restrict__ alpha, float* amax1) {
  int e = blockIdx.x * blockDim.x + threadIdx.x;
  if (e >= E_TOT) return;
  int s, d; edge_endpoints(ei, e, s, d);
  float a = consts[0] * x[s] + consts[1] * x[d];
  a = (a > 0.f) ? a : NEG_SLOPE * a;
  alpha[e] = a;
  atomicMaxFloat(&amax1[d], a);
}

// ---------- layer 1, pass B: e=exp(a-amax); denom += e; numer += e*x[src] (scalar agg) ----------
__global__ __launch_bounds__(256) void l1_edgeB(const float* __restrict__ x, const int* __restrict__ ei,
                                                const float* __restrict__ alpha, const float* __restrict__ amax1,
                                                float* denom1, float* numer1) {
  int e = blockIdx.x * blockDim.x + threadIdx.x;
  if (e >= E_TOT) return;
  int s, d; edge_endpoints(ei, e, s, d);
  float ee = __expf(alpha[e] - amax1[d]);
  atomicAdd(&denom1[d], ee);
  atomicAdd(&numer1[d], ee * x[s]);
}

// ---------- layer 1, pass C: h1[n][f] = relu(W1[f]*numer/denom + b1[f]) ----------
__global__ __launch_bounds__(256) void l1_nodeC(const float* __restrict__ numer1, const float* __restrict__ denom1,
                                                const float* __restrict__ W1, const float* __restrict__ b1,
                                                float* __restrict__ h1) {
  int i = blockIdx.x * blockDim.x + threadIdx.x;        // N_NODES*32
  if (i >= N_NODES * 32) return;
  int n = i >> 5, f = i & 31;
  float t = numer1[n] / (denom1[n] + EPS_SM);
  float h = W1[f] * t + b1[f];
  h1[i] = fmaxf(h, 0.f);
}

// ---------- layer 2 GEMM: h2pre[100000,64] = h1[100000,32] @ W2[32,64] via V_WMMA_F32_16X16X4_F32 ----------
// One wave per 16x16 output tile; 4 waves/block cover the 4 N-tiles of one M-tile (A rows shared in cache).
__global__ __launch_bounds__(128) void gat_gemm_wmma(const float* __restrict__ h1,
                                                     const float* __restrict__ W2,
                                                     float* __restrict__ h2) {
  const int lane = threadIdx.x & 31;
  const int wave = threadIdx.x >> 5;     // N-tile 0..3
  const int m0   = blockIdx.x << 4;      // 100000 = 16*6250, no tail
  const int n0   = wave << 4;
  const int r    = lane & 15;
  const int hsel = lane >> 4;            // lane-half selects K pair / M upper half
  v8f acc = {};
#if __has_builtin(__builtin_amdgcn_wmma_f32_16x16x4_f32)
  #pragma unroll
  for (int k0 = 0; k0 < 32; k0 += 4) {
    const int ka = k0 + (hsel << 1);     // A: VGPR0=K0|K2, VGPR1=K1|K3 by lane half
    v2f a, b;
    a[0] = h1[(m0 + r) * 32 + ka + 0];
    a[1] = h1[(m0 + r) * 32 + ka + 1];
    b[0] = W2[(ka + 0) * 64 + n0 + r];   // B rows striped across lanes, same half pattern
    b[1] = W2[(ka + 1) * 64 + n0 + r];
    acc = __builtin_amdgcn_wmma_f32_16x16x4_f32(false, a, false, b, (short)0, acc,
                                                false, false);
  }
#else
  // exact scalar fallback matching the D-store mapping below
  #pragma unroll
  for (int j = 0; j < 8; ++j) {
    float sum = 0.f;
    const int row = m0 + j + (hsel << 3);
    for (int k = 0; k < 32; ++k) sum += h1[row * 32 + k] * W2[k * 64 + n0 + r];
    acc[j] = sum;
  }
#endif
  #pragma unroll
  for (int j = 0; j < 8; ++j) {          // C/D: VGPR j -> rows j (lanes 0-15) / j+8 (lanes 16-31)
    h2[(m0 + j + (hsel << 3)) * 64 + n0 + r] = acc[j];
  }
}

// ---------- per-node attention scalars: s2 = h2pre·a2s, d2 = h2pre·a2d ----------
__global__ __launch_bounds__(256) void l2_sd(const float* __restrict__ h2pre,
                                             const float* __restrict__ a2s, const float* __restrict__ a2d,
                                             float* __restrict__ s2, float* __restrict__ d2) {
  int n = blockIdx.x * blockDim.x + threadIdx.x;
  if (n >= N_NODES) return;
  const float4* hv = (const float4*)(h2pre + (size_t)n * 64);
  float ss = 0.f, dd = 0.f;
  #pragma unroll
  for (int q = 0; q < 16; ++q) {
    float4 h = hv[q];
    float4 as = ((const float4*)a2s)[q];
    float4 ad = ((const float4*)a2d)[q];
    ss += h.x * as.x + h.y * as.y + h.z * as.z + h.w * as.w;
    dd += h.x * ad.x + h.y * ad.y + h.z * ad.z + h.w * ad.w;
  }
  s2[n] = ss; d2[n] = dd;
}

// ---------- layer 2, pass A ----------
__global__ __launch_bounds__(256) void l2_edgeA(const int* __restrict__ ei,
                                                const float* __restrict__ s2, const float* __restrict__ d2,
                                                float* __restrict__ alpha, float* amax2) {
  int e = blockIdx.x * blockDim.x + threadIdx.x;
  if (e >= E_TOT) return;
  int s, d; edge_endpoints(ei, e, s, d);
  float a = s2[s] + d2[d];
  a = (a > 0.f) ? a : NEG_SLOPE * a;
  alpha[e] = a;
  atomicMaxFloat(&amax2[d], a);
}

// ---------- layer 2, pass B: 16 threads per edge, float4 row slices; L2-resident scatter ----------
__global__ __launch_bounds__(256) void l2_edgeB(const int* __restrict__ ei,
                                                const float* __restrict__ alpha, const float* __restrict__ amax2,
                                                const float* __restrict__ h2pre,
                                                float* denom2, float* numer2) {
  int t = blockIdx.x * blockDim.x + threadIdx.x;        // E_TOT*16 threads
  int e = t >> 4;
  if (e >= E_TOT) return;
  int fo = (t & 15) << 2;
  int s, d; edge_endpoints(ei, e, s, d);
  float ee = __expf(alpha[e] - amax2[d]);
  if (fo == 0) atomicAdd(&denom2[d], ee);
  float4 h = *(const float4*)(h2pre + (size_t)s * 64 + fo);
  float* dst = numer2 + (size_t)d * 64 + fo;
  atomicAdd(dst + 0, ee * h.x);
  atomicAdd(dst + 1, ee * h.y);
  atomicAdd(dst + 2, ee * h.z);
  atomicAdd(dst + 3, ee * h.w);
}

// ---------- layer 2, pass C + mean-pool accumulation ----------
__global__ __launch_bounds__(256) void l2_nodeC_pool(const float* __restrict__ numer2,
                                                     const float* __restrict__ denom2,
                                                     const float* __restrict__ b2,
                                                     const int* __restrict__ batch,
                                                     float* pooled, float* cnt) {
  int i = blockIdx.x * blockDim.x + threadIdx.x;        // N_NODES*64
  if (i >= N_NODES * 64) return;
  int n = i >> 6, f = i & 63;
  float v = numer2[i] / (denom2[n] + EPS_SM) + b2[f];
  v = fmaxf(v, 0.f);
  int g = batch[n];
  atomicAdd(&pooled[g * 64 + f], v);
  if (f == 0) atomicAdd(&cnt[g], 1.0f);
}

// ---------- final FC: out[g,c] = (pooled[g]/max(cnt,1)) · fcW[:,c] + fcb[c] ----------
__global__ __launch_bounds__(256) void gat_final(const float* __restrict__ pooled, const float* __restrict__ cnt,
                                                 const float* __restrict__ fcW, const float* __restrict__ fcb,
                                                 float* __restrict__ out) {
  int i = blockIdx.x * blockDim.x + threadIdx.x;        // 512
  if (i >= N_GRAPHS * 2) return;
  int g = i >> 1, c = i & 1;
  float cn = fmaxf(cnt[g], 1.0f);
  float acc = fcb[c];
  for (int f = 0; f < 64; ++f) acc += (pooled[g * 64 + f] / cn) * fcW[f * 2 + c];
  out[i] = acc;
}

extern "C" void kernel_launch(void* const* d_in, const int* in_sizes, int n_in,
                              void* d_out, int out_size, void* d_ws, size_t ws_size,
                              hipStream_t stream) {
  const float* x    = (const float*)d_in[0];
  const int*   ei   = (const int*)  d_in[1];
  const int*   batch= (const int*)  d_in[2];
  const float* W1   = (const float*)d_in[3];
  const float* a1s  = (const float*)d_in[4];
  const float* a1d  = (const float*)d_in[5];
  const float* b1   = (const float*)d_in[6];
  const float* W2   = (const float*)d_in[7];
  const float* a2s  = (const float*)d_in[8];
  const float* a2d  = (const float*)d_in[9];
  const float* b2   = (const float*)d_in[10];
  const float* fcW  = (const float*)d_in[11];
  const float* fcb  = (const float*)d_in[12];
  float* out = (float*)d_out;

  // workspace carve-up (floats); every offset is 16B-aligned. Total ~18.42M floats (~74MB).
  float* ws = (float*)d_ws;
  size_t o = 0;
  float* consts = ws + o; o += 8;
  float* amax1  = ws + o; o += N_NODES;
  float* denom1 = ws + o; o += N_NODES;
  float* numer1 = ws + o; o += N_NODES;
  float* alpha  = ws + o; o += E_TOT;               // reused by both layers
  float* h1     = ws + o; o += (size_t)N_NODES * 32;
  float* h2pre  = ws + o; o += (size_t)N_NODES * 64;
  float* s2     = ws + o; o += N_NODES;
  float* d2     = ws + o; o += N_NODES;
  float* amax2  = ws + o; o += N_NODES;
  float* denom2 = ws + o; o += N_NODES;
  float* numer2 = ws + o; o += (size_t)N_NODES * 64;
  float* pooled = ws + o; o += N_GRAPHS * 64;
  float* cnt    = ws + o; o += N_GRAPHS;

  const int B = 256;
  const int gInit  = (N_NODES * 64 + B - 1) / B;
  const int gEdge  = (E_TOT + B - 1) / B;
  const int gN32   = (N_NODES * 32 + B - 1) / B;
  const int gNode  = (N_NODES + B - 1) / B;
  const int gEdge16= (E_TOT * 16 + B - 1) / B;
  const int gN64   = (N_NODES * 64 + B - 1) / B;

  gat_init<<<gInit, B, 0, stream>>>(amax1, denom1, numer1, amax2, denom2, numer2, pooled, cnt);
  gat_prep<<<1, 32, 0, stream>>>(W1, a1s, a1d, consts);

  // layer 1 (attention collapses to scalars since Fin==1)
  l1_edgeA<<<gEdge, B, 0, stream>>>(x, ei, consts, alpha, amax1);
  l1_edgeB<<<gEdge, B, 0, stream>>>(x, ei, alpha, amax1, denom1, numer1);
  l1_nodeC<<<gN32, B, 0, stream>>>(numer1, denom1, W1, b1, h1);

  // layer 2
  gat_gemm_wmma<<<N_NODES / 16, 128, 0, stream>>>(h1, W2, h2pre);
  l2_sd<<<gNode, B, 0, stream>>>(h2pre, a2s, a2d, s2, d2);
  l2_edgeA<<<gEdge, B, 0, stream>>>(ei, s2, d2, alpha, amax2);
  l2_edgeB<<<gEdge16, B, 0, stream>>>(ei, alpha, amax2, h2pre, denom2, numer2);
  l2_nodeC_pool<<<gN64, B, 0, stream>>>(numer2, denom2, b2, batch, pooled, cnt);

  // readout
  gat_final<<<2, B, 0, stream>>>(pooled, cnt, fcW, fcb, out);
}